// CD_GCN_7327214207509
// MI455X (gfx1250) — compile-verified
//
#include <hip/hip_runtime.h>

// ---------------------------------------------------------------------------
// Problem constants (from reference): T=12, N=20000, E=640000, F_IN=128, H=64
// ---------------------------------------------------------------------------
#define T_DIM 12
#define N_DIM 20000
#define E_DIM 640000
#define F_DIM 128
#define H_DIM 64
#define G_DIM 256   // 4*H gates

typedef __attribute__((ext_vector_type(2))) float v2f;
typedef __attribute__((ext_vector_type(8))) float v8f;

#if defined(__HIP_DEVICE_COMPILE__) && !__has_builtin(__builtin_amdgcn_wmma_f32_16x16x4_f32)
#error "gfx1250 device pass: __builtin_amdgcn_wmma_f32_16x16x4_f32 not available"
#endif

__device__ __forceinline__ v8f wmma_f32x4(v2f a, v2f b, v8f c) {
    // D = A(16x4 f32) * B(4x16 f32) + C(16x16 f32); exact fp32 math.
    return __builtin_amdgcn_wmma_f32_16x16x4_f32(
        /*neg_a=*/false, a, /*neg_b=*/false, b,
        /*c_mod=*/(short)0, c, /*reuse_a=*/false, /*reuse_b=*/false);
}

__device__ __forceinline__ float sigf(float x) {
    return 1.0f / (1.0f + __expf(-x));
}

// ---------------------------------------------------------------------------
// k0: init deg=1.0 (self loop), h_state=0, c_state=0
// ---------------------------------------------------------------------------
__global__ void k_init(float* __restrict__ deg, float* __restrict__ hstate,
                       float* __restrict__ cstate) {
    long i = (long)blockIdx.x * blockDim.x + threadIdx.x;
    const long ndeg = (long)T_DIM * N_DIM;
    const long nst  = (long)N_DIM * H_DIM;
    if (i < ndeg) {
        deg[i] = 1.0f;
    } else if (i < ndeg + nst) {
        hstate[i - ndeg] = 0.0f;
        cstate[i - ndeg] = 0.0f;
    }
}

// ---------------------------------------------------------------------------
// k1: h_lin[t,n,:] = x[t,n,:] @ W_gcn   (fp32 WMMA, K=128)
//     grid.x = (T*N)/16 blocks, 128 threads = 4 waves, one 16-col tile each.
// ---------------------------------------------------------------------------
__global__ void k_gemm_gcn(const float* __restrict__ x,
                           const float* __restrict__ Wg,
                           float* __restrict__ hlin) {
    __shared__ float At[16 * 129];     // 16 rows x 128 cols, padded
    const int rowBase = blockIdx.x * 16;
    const int tid = threadIdx.x;

    // Cooperative coalesced load of the A tile (float4).
    for (int i = tid; i < 16 * 32; i += 128) {
        int r  = i >> 5;
        int c4 = i & 31;
        float4 v = ((const float4*)(x + (size_t)(rowBase + r) * F_DIM))[c4];
        float* p = &At[r * 129 + c4 * 4];
        p[0] = v.x; p[1] = v.y; p[2] = v.z; p[3] = v.w;
    }
    __syncthreads();

    const int wave  = tid >> 5;         // 0..3 -> H column tile
    const int lane  = tid & 31;
    const int m     = lane & 15;
    const int kk    = (lane >> 4) * 2;  // K sub-offset per half-wave
    const int n     = lane & 15;
    const int jbase = wave * 16;

    v8f acc = {};
    for (int k = 0; k < F_DIM / 4; ++k) {
        const int kc = k * 4 + kk;
        v2f a, b;
        a.x = At[m * 129 + kc];
        a.y = At[m * 129 + kc + 1];
        b.x = Wg[(size_t)kc * H_DIM + jbase + n];
        b.y = Wg[(size_t)(kc + 1) * H_DIM + jbase + n];
        acc = wmma_f32x4(a, b, acc);
    }

    // D layout: VGPR r, lanes 0-15 -> row r; lanes 16-31 -> row 8+r.
    float* outp = hlin + (size_t)(rowBase + (lane >> 4) * 8) * H_DIM + jbase + (lane & 15);
#pragma unroll
    for (int r = 0; r < 8; ++r)
        outp[(size_t)r * H_DIM] = acc[r];
}

// ---------------------------------------------------------------------------
// k2: deg[t, dst] += 1 for every edge (float atomics, L2-resident)
// ---------------------------------------------------------------------------
__global__ void k_deg(const int* __restrict__ ei, float* __restrict__ deg) {
    long i = (long)blockIdx.x * blockDim.x + threadIdx.x;   // [0, T*E)
    int t = (int)(i / E_DIM);
    int e = (int)(i - (long)t * E_DIM);
    int d = ei[(size_t)t * 2 * E_DIM + E_DIM + e];
    atomicAdd(&deg[(size_t)t * N_DIM + d], 1.0f);
}

// ---------------------------------------------------------------------------
// k3: self-loop init: agg[t,n,c] = h_lin[t,n,c] * (dinv[n]^2) = h_lin / deg
// ---------------------------------------------------------------------------
__global__ void k_selfinit(const float* __restrict__ hlin,
                           const float* __restrict__ deg,
                           float* __restrict__ agg) {
    long i = (long)blockIdx.x * blockDim.x + threadIdx.x;   // [0, T*N*H)
    long node = i >> 6;
    agg[i] = hlin[i] / deg[node];
}

// ---------------------------------------------------------------------------
// k4: edge scatter. One wave per edge; 2 channels/lane; coalesced 128B
//     gather + fp32 atomic scatter (per-t working set ~10MB -> lives in L2).
// ---------------------------------------------------------------------------
__global__ void k_scatter(const int* __restrict__ ei,
                          const float* __restrict__ hlin,
                          const float* __restrict__ deg,
                          float* __restrict__ agg) {
    long eg = ((long)blockIdx.x * blockDim.x + threadIdx.x) >> 5;  // edge id in [0, T*E)
    int t = (int)(eg / E_DIM);
    int e = (int)(eg - (long)t * E_DIM);
    const int* base = ei + (size_t)t * 2 * E_DIM;
    int s = base[e];
    int d = base[E_DIM + e];
    float norm = __frsqrt_rn(deg[(size_t)t * N_DIM + s]) *
                 __frsqrt_rn(deg[(size_t)t * N_DIM + d]);
    const float* hp = hlin + ((size_t)t * N_DIM + s) * H_DIM;
    float* ap       = agg  + ((size_t)t * N_DIM + d) * H_DIM;
    int lane = threadIdx.x & 31;
    atomicAdd(&ap[lane],      hp[lane]      * norm);
    atomicAdd(&ap[lane + 32], hp[lane + 32] * norm);
}

// ---------------------------------------------------------------------------
// k5: gcn_out = relu(agg + b_gcn)  (in place)
// ---------------------------------------------------------------------------
__global__ void k_biasrelu(float* __restrict__ agg, const float* __restrict__ bg) {
    long i = (long)blockIdx.x * blockDim.x + threadIdx.x;   // [0, T*N*H)
    int c = (int)(i & (H_DIM - 1));
    float v = agg[i] + bg[c];
    agg[i] = v > 0.0f ? v : 0.0f;
}

// ---------------------------------------------------------------------------
// k6: one LSTM time step. Block = 512 thr = 16 waves; owns 16 nodes x 256
//     gate cols. gates = seq_t @ Wih^T + h_prev @ Whh^T  via fp32 WMMA (K=64),
//     staged through LDS for the cross-tile i/f/g/o nonlinearity.
// ---------------------------------------------------------------------------
__global__ void k_lstm_step(const float* __restrict__ seq,    // gcn_out + t*N*H
                            const float* __restrict__ Wih,    // [256,64] row-major
                            const float* __restrict__ Whh,    // [256,64]
                            const float* __restrict__ bih,
                            const float* __restrict__ bhh,
                            float* __restrict__ hstate,
                            float* __restrict__ cstate,
                            float* __restrict__ out) {        // d_out + t*N*H
    __shared__ float Ax[16 * 65];
    __shared__ float Ah[16 * 65];
    __shared__ float G[16 * 257];      // 16 rows x 256 gate cols, padded

    const int rowBase = blockIdx.x * 16;
    const int tid = threadIdx.x;

    for (int i = tid; i < 16 * H_DIM; i += 512) {
        int r = i >> 6, c = i & 63;
        Ax[r * 65 + c] = seq[(size_t)(rowBase + r) * H_DIM + c];
        Ah[r * 65 + c] = hstate[(size_t)(rowBase + r) * H_DIM + c];
    }
    __syncthreads();

    const int wave  = tid >> 5;          // 0..15 -> gate column tile
    const int lane  = tid & 31;
    const int m     = lane & 15;
    const int kk    = (lane >> 4) * 2;
    const int n     = lane & 15;
    const int jbase = wave * 16;

    v8f acc = {};
    for (int k = 0; k < H_DIM / 4; ++k) {
        const int kc = k * 4 + kk;
        v2f a, b;
        // input contribution: B[k][j] = Wih[j*64 + k]
        a.x = Ax[m * 65 + kc];
        a.y = Ax[m * 65 + kc + 1];
        b.x = Wih[(size_t)(jbase + n) * H_DIM + kc];
        b.y = Wih[(size_t)(jbase + n) * H_DIM + kc + 1];
        acc = wmma_f32x4(a, b, acc);
        // recurrent contribution
        a.x = Ah[m * 65 + kc];
        a.y = Ah[m * 65 + kc + 1];
        b.x = Whh[(size_t)(jbase + n) * H_DIM + kc];
        b.y = Whh[(size_t)(jbase + n) * H_DIM + kc + 1];
        acc = wmma_f32x4(a, b, acc);
    }

    const int grow = (lane >> 4) * 8;
#pragma unroll
    for (int r = 0; r < 8; ++r)
        G[(grow + r) * 257 + jbase + (lane & 15)] = acc[r];
    __syncthreads();

    // Elementwise gate math: 16x64 elements, 512 threads -> 2 each.
    for (int i = tid; i < 16 * H_DIM; i += 512) {
        int r = i >> 6, h = i & 63;
        size_t node = (size_t)(rowBase + r);
        float gi = G[r * 257 + h]             + bih[h]             + bhh[h];
        float gf = G[r * 257 + H_DIM + h]     + bih[H_DIM + h]     + bhh[H_DIM + h];
        float gg = G[r * 257 + 2 * H_DIM + h] + bih[2 * H_DIM + h] + bhh[2 * H_DIM + h];
        float go = G[r * 257 + 3 * H_DIM + h] + bih[3 * H_DIM + h] + bhh[3 * H_DIM + h];
        float cp = cstate[node * H_DIM + h];
        float cn = sigf(gf) * cp + sigf(gi) * tanhf(gg);
        float hn = sigf(go) * tanhf(cn);
        cstate[node * H_DIM + h] = cn;
        hstate[node * H_DIM + h] = hn;
        out[node * H_DIM + h]    = hn;
    }
}

// ---------------------------------------------------------------------------
extern "C" void kernel_launch(void* const* d_in, const int* in_sizes, int n_in,
                              void* d_out, int out_size, void* d_ws, size_t ws_size,
                              hipStream_t stream) {
    (void)in_sizes; (void)n_in; (void)out_size; (void)ws_size;

    const float* x   = (const float*)d_in[0];
    const int*   ei  = (const int*)d_in[1];
    const float* Wg  = (const float*)d_in[2];
    const float* bg  = (const float*)d_in[3];
    const float* Wih = (const float*)d_in[4];
    const float* Whh = (const float*)d_in[5];
    const float* bih = (const float*)d_in[6];
    const float* bhh = (const float*)d_in[7];
    float* out = (float*)d_out;

    // Workspace layout (floats): hlin | agg | deg | hstate | cstate
    float* ws     = (float*)d_ws;
    float* hlin   = ws;
    float* agg    = hlin + (size_t)T_DIM * N_DIM * H_DIM;
    float* deg    = agg  + (size_t)T_DIM * N_DIM * H_DIM;
    float* hstate = deg  + (size_t)T_DIM * N_DIM;
    float* cstate = hstate + (size_t)N_DIM * H_DIM;

    const long initN = (long)T_DIM * N_DIM + (long)N_DIM * H_DIM;  // deg + state
    k_init<<<(unsigned)((initN + 255) / 256), 256, 0, stream>>>(deg, hstate, cstate);

    k_gemm_gcn<<<(T_DIM * N_DIM) / 16, 128, 0, stream>>>(x, Wg, hlin);

    k_deg<<<((long)T_DIM * E_DIM) / 256, 256, 0, stream>>>(ei, deg);

    k_selfinit<<<((long)T_DIM * N_DIM * H_DIM) / 256, 256, 0, stream>>>(hlin, deg, agg);

    k_scatter<<<((long)T_DIM * E_DIM) / 8, 256, 0, stream>>>(ei, hlin, deg, agg);

    k_biasrelu<<<((long)T_DIM * N_DIM * H_DIM) / 256, 256, 0, stream>>>(agg, bg);

    for (int t = 0; t < T_DIM; ++t) {
        k_lstm_step<<<N_DIM / 16, 512, 0, stream>>>(
            agg + (size_t)t * N_DIM * H_DIM, Wih, Whh, bih, bhh,
            hstate, cstate, out + (size_t)t * N_DIM * H_DIM);
    }
}